// AdditiveAttention_26053271617960
// MI455X (gfx1250) — compile-verified
//
#include <hip/hip_runtime.h>
#include <hip/hip_bf16.h>

// Problem constants (match reference)
constexpr int BB = 2;      // batch
constexpr int LQN = 1024;  // Lq
constexpr int LKN = 1024;  // Lk
constexpr int DN = 64;     // feature dim

typedef float v2f __attribute__((ext_vector_type(2)));
typedef float v8f __attribute__((ext_vector_type(8)));

__device__ __forceinline__ float fast_tanh(float x) {
#if __has_builtin(__builtin_amdgcn_tanhf)
    return __builtin_amdgcn_tanhf(x);   // gfx1250 v_tanh_f32
#else
    // tanh(x) = 1 - 2/(exp(2x)+1); exp2/rcp are native trans ops
    float e = __builtin_amdgcn_exp2f(x * 2.8853900817779268f); // 2*log2(e)
    return 1.0f - 2.0f * __builtin_amdgcn_rcpf(e + 1.0f);
#endif
}

// y[r][n] = sum_d x[r][d] * W[n][d] + bias[n]   (x: rows x 64, W: 64 x 64)
// One wave computes a 16-row x 64-col output tile using V_WMMA_F32_16X16X4_F32.
__global__ __launch_bounds__(256) void proj_wmma_kernel(
    const float* __restrict__ x, const float* __restrict__ W,
    const float* __restrict__ bias, float* __restrict__ y)
{
    const int wave = blockIdx.x * 8 + (threadIdx.x >> 5);
    const int lane = threadIdx.x & 31;
    const int row0 = wave * 16;
    const int m    = lane & 15;   // A-row / B-col / D-col index
    const int half = lane >> 4;   // lane half selects K pair (A/B) or M+8 (C/D)

    v8f acc0 = {}, acc1 = {}, acc2 = {}, acc3 = {};
    const float* xrow = x + (size_t)(row0 + m) * DN;

    for (int kc = 0; kc < 16; ++kc) {          // K = 64 in chunks of 4
        const int kb = kc * 4 + 2 * half;      // K index for vgpr component 0
        v2f a;                                 // A 16x4: lane holds K = kb, kb+1
        a.x = xrow[kb];
        a.y = xrow[kb + 1];
        // B 4x16: element B[k][n] = W[n][k]; lane holds K = kb+v, N = m + 16*nt
        v2f b0, b1, b2, b3;
        b0.x = W[(m +  0) * DN + kb]; b0.y = W[(m +  0) * DN + kb + 1];
        b1.x = W[(m + 16) * DN + kb]; b1.y = W[(m + 16) * DN + kb + 1];
        b2.x = W[(m + 32) * DN + kb]; b2.y = W[(m + 32) * DN + kb + 1];
        b3.x = W[(m + 48) * DN + kb]; b3.y = W[(m + 48) * DN + kb + 1];
        acc0 = __builtin_amdgcn_wmma_f32_16x16x4_f32(false, a, false, b0, (short)0, acc0, false, false);
        acc1 = __builtin_amdgcn_wmma_f32_16x16x4_f32(false, a, false, b1, (short)0, acc1, false, false);
        acc2 = __builtin_amdgcn_wmma_f32_16x16x4_f32(false, a, false, b2, (short)0, acc2, false, false);
        acc3 = __builtin_amdgcn_wmma_f32_16x16x4_f32(false, a, false, b3, (short)0, acc3, false, false);
    }

    const float bv0 = bias[m], bv1 = bias[m + 16], bv2 = bias[m + 32], bv3 = bias[m + 48];
#pragma unroll
    for (int v = 0; v < 8; ++v) {              // D layout: M = v + 8*half, N = m + 16*nt
        const size_t rm = (size_t)(row0 + v + 8 * half) * DN;
        y[rm + m +  0] = acc0[v] + bv0;
        y[rm + m + 16] = acc1[v] + bv1;
        y[rm + m + 32] = acc2[v] + bv2;
        y[rm + m + 48] = acc3[v] + bv3;
    }
}

// scores[b,q,k] = b_score + sum_d w[d] * tanh(qt[b,q,d] + kt[b,k,d])
// Block: 256 threads, tile 32(q) x 128(k); each thread owns a 2x8 micro-tile.
__global__ __launch_bounds__(256) void score_kernel(
    const float* __restrict__ qt, const float* __restrict__ kt,
    const float* __restrict__ w, const float* __restrict__ bsc,
    float* __restrict__ out)
{
    constexpr int TQ = 32, TK = 128, PAD = 65; // 65-float row pitch vs 64 LDS banks
    __shared__ float qs[TQ * PAD];
    __shared__ float ks[TK * PAD];
    __shared__ float ws[DN];

    const int t  = threadIdx.x;
    const int b  = blockIdx.z;
    const int q0 = blockIdx.y * TQ;
    const int k0 = blockIdx.x * TK;

    const float* qbase = qt + ((size_t)b * LQN + q0) * DN;
    const float* kbase = kt + ((size_t)b * LKN + k0) * DN;

#pragma unroll
    for (int i = 0; i < 8; ++i) {              // 32*64 floats, coalesced
        const int idx = t + i * 256;
        qs[(idx >> 6) * PAD + (idx & 63)] = qbase[idx];
    }
#pragma unroll
    for (int i = 0; i < 32; ++i) {             // 128*64 floats, coalesced
        const int idx = t + i * 256;
        ks[(idx >> 6) * PAD + (idx & 63)] = kbase[idx];
    }
    if (t < DN) ws[t] = w[t];
    __syncthreads();

    const int kg = t & 15;                     // 8 consecutive k columns
    const int qp = t >> 4;                     // 2 consecutive q rows
    const float* qrow0 = &qs[(qp * 2 + 0) * PAD];
    const float* qrow1 = &qs[(qp * 2 + 1) * PAD];
    const float* krow  = &ks[(kg * 8) * PAD];

    float acc0[8], acc1[8];
#pragma unroll
    for (int j = 0; j < 8; ++j) { acc0[j] = 0.0f; acc1[j] = 0.0f; }

#pragma unroll 4
    for (int d = 0; d < DN; ++d) {
        const float wv = ws[d];
        const float qa = qrow0[d];
        const float qb = qrow1[d];
#pragma unroll
        for (int j = 0; j < 8; ++j) {
            const float kv = krow[j * PAD + d];
            acc0[j] += wv * fast_tanh(qa + kv);
            acc1[j] += wv * fast_tanh(qb + kv);
        }
    }

    const float bias = bsc[0];
    const size_t obase = ((size_t)b * LQN + (q0 + qp * 2)) * (size_t)LKN + k0 + kg * 8;
#pragma unroll
    for (int j = 0; j < 8; ++j) {
        out[obase + j]        = acc0[j] + bias;
        out[obase + LKN + j]  = acc1[j] + bias;
    }
}

extern "C" void kernel_launch(void* const* d_in, const int* in_sizes, int n_in,
                              void* d_out, int out_size, void* d_ws, size_t ws_size,
                              hipStream_t stream) {
    const float* q_in    = (const float*)d_in[0]; // [B,LQ,D]
    const float* k_in    = (const float*)d_in[1]; // [B,LK,D]
    const float* Wq      = (const float*)d_in[2]; // [D,D]
    const float* bq      = (const float*)d_in[3]; // [D]
    const float* Wk      = (const float*)d_in[4]; // [D,D]
    const float* bk      = (const float*)d_in[5]; // [D]
    const float* w_score = (const float*)d_in[6]; // [D]
    const float* b_score = (const float*)d_in[7]; // [1]
    float* out = (float*)d_out;                   // [B,LQ,LK]

    // Workspace: qt (2048*64 f32) + kt (2048*64 f32) = 1 MiB
    float* qt = (float*)d_ws;
    float* kt = qt + (size_t)BB * LQN * DN;

    // 2048 rows / (16 rows per wave * 8 waves per block) = 16 blocks each
    proj_wmma_kernel<<<(BB * LQN) / 128, 256, 0, stream>>>(q_in, Wq, bq, qt);
    proj_wmma_kernel<<<(BB * LKN) / 128, 256, 0, stream>>>(k_in, Wk, bk, kt);

    dim3 grid(LKN / 128, LQN / 32, BB);
    score_kernel<<<grid, 256, 0, stream>>>(qt, kt, w_score, b_score, out);
}